// GraphAttentionLayer_8246337208369
// MI455X (gfx1250) — compile-verified
//
#include <hip/hip_runtime.h>
#include <hip/hip_bf16.h>

typedef float v2f __attribute__((ext_vector_type(2)));
typedef float v8f __attribute__((ext_vector_type(8)));

#define NNODES 2048
#define FDIM   128
#define NBATCH 16

// D = A(16x4 f32) * B(4x16 f32) + C(16x16 f32)  -- CDNA5 V_WMMA_F32_16X16X4_F32
__device__ __forceinline__ v8f wmma_f32_16x16x4(v2f a, v2f b, v8f c) {
  return __builtin_amdgcn_wmma_f32_16x16x4_f32(
      /*neg_a=*/false, a, /*neg_b=*/false, b,
      /*c_mod=*/(short)0, c, /*reuse_a=*/false, /*reuse_b=*/false);
}

__device__ __forceinline__ float lrelu(float x) {
  return fmaxf(x, 0.0f) + 0.01f * fminf(x, 0.0f);
}

// Generic pointer to a __shared__ object: LDS aperture occupies addr[63:32],
// so the low 32 bits are exactly the LDS byte offset the async ops want.
__device__ __forceinline__ uint32_t lds_byte_off(const void* p) {
  return (uint32_t)(uintptr_t)p;
}

// CDNA5 async copy global -> LDS (ASYNCcnt-tracked, no VGPR data movement).
__device__ __forceinline__ void async_copy_b128(uint32_t lds_off, const void* gptr) {
  asm volatile("global_load_async_to_lds_b128 %0, %1, off"
               :
               : "v"(lds_off), "v"((unsigned long long)(uintptr_t)gptr)
               : "memory");
}
__device__ __forceinline__ void async_copy_b32(uint32_t lds_off, const void* gptr) {
  asm volatile("global_load_async_to_lds_b32 %0, %1, off"
               :
               : "v"(lds_off), "v"((unsigned long long)(uintptr_t)gptr)
               : "memory");
}
__device__ __forceinline__ void wait_async0() {
  asm volatile("s_wait_asynccnt 0" ::: "memory");
}

// ---------------------------------------------------------------------------
// Kernel 1: Wh[b] = h[b] @ W    (2048x128 @ 128x128, per batch)
// grid = (16 row-tiles, 16 batches), block = 256 (8 waves); wave -> 16 rows x 128 cols
// ---------------------------------------------------------------------------
__global__ __launch_bounds__(256) void k_wh_gemm(const float* __restrict__ h,
                                                 const float* __restrict__ W,
                                                 float* __restrict__ Wh) {
  __shared__ float Ws[FDIM * FDIM];  // 64 KB of the 320 KB WGP LDS
  const int tid = threadIdx.x;
  const int b = blockIdx.y;

  for (int i = tid * 4; i < FDIM * FDIM; i += 256 * 4)
    *(float4*)&Ws[i] = *(const float4*)&W[i];
  __syncthreads();

  const int wave = tid >> 5, lane = tid & 31;
  const int ln = lane & 15, hi = lane >> 4;
  const int r0 = blockIdx.x * 128 + wave * 16;
  const float* hrow = h + ((size_t)b * NNODES + r0 + ln) * FDIM;

  v8f acc[8] = {};
#pragma unroll 4
  for (int k0 = 0; k0 < FDIM; k0 += 4) {
    // A frag: lanes0-15 -> (M=ln, K=k0..k0+1), lanes16-31 -> (M=ln, K=k0+2..k0+3)
    v2f afrag = *(const v2f*)&hrow[k0 + hi * 2];
#pragma unroll
    for (int c = 0; c < 8; ++c) {
      v2f bfrag;
      bfrag.x = Ws[(k0 + hi * 2 + 0) * FDIM + c * 16 + ln];
      bfrag.y = Ws[(k0 + hi * 2 + 1) * FDIM + c * 16 + ln];
      acc[c] = wmma_f32_16x16x4(afrag, bfrag, acc[c]);
    }
  }

  float* ob = Wh + (size_t)b * NNODES * FDIM;
#pragma unroll
  for (int c = 0; c < 8; ++c)
#pragma unroll
    for (int v = 0; v < 8; ++v)
      ob[(size_t)(r0 + v + hi * 8) * FDIM + c * 16 + ln] = acc[c][v];
}

// ---------------------------------------------------------------------------
// Kernel 2: f1 = Wh @ a1 ; f2 = Wh @ a2   (one thread per row)
// ---------------------------------------------------------------------------
__global__ __launch_bounds__(256) void k_f1f2(const float* __restrict__ Wh,
                                              const float* __restrict__ a,
                                              float* __restrict__ f1,
                                              float* __restrict__ f2) {
  __shared__ float a1s[FDIM], a2s[FDIM];
  const int tid = threadIdx.x;
  if (tid < FDIM) { a1s[tid] = a[tid]; a2s[tid] = a[FDIM + tid]; }
  __syncthreads();
  const size_t row = (size_t)blockIdx.x * 256 + tid;  // over B*N rows
  const float* wr = Wh + row * FDIM;
  float s1 = 0.f, s2 = 0.f;
#pragma unroll 8
  for (int f = 0; f < FDIM; ++f) {
    float w = wr[f];
    s1 = fmaf(w, a1s[f], s1);
    s2 = fmaf(w, a2s[f], s2);
  }
  f1[row] = s1;
  f2[row] = s2;
}

// ---------------------------------------------------------------------------
// Kernel 3: per-column softmax stats (softmax over axis=1 == over i)
//   m[b,j] = max_i lrelu(f1[b,i]+f2[b,j]);  r[b,j] = 1 / sum_i exp(x - m)
// ---------------------------------------------------------------------------
__global__ __launch_bounds__(256) void k_softmax_stats(const float* __restrict__ f1,
                                                       const float* __restrict__ f2,
                                                       float* __restrict__ mcol,
                                                       float* __restrict__ rcol) {
  __shared__ float f1s[NNODES];  // 8 KB
  const int b = blockIdx.y, tid = threadIdx.x;
  for (int i = tid; i < NNODES; i += 256) f1s[i] = f1[(size_t)b * NNODES + i];
  __syncthreads();

  const int j = blockIdx.x * 256 + tid;
  const float fj = f2[(size_t)b * NNODES + j];
  float mx = -3.402823e38f;
#pragma unroll 4
  for (int i = 0; i < NNODES; ++i) mx = fmaxf(mx, lrelu(f1s[i] + fj));
  float d = 0.f;
#pragma unroll 4
  for (int i = 0; i < NNODES; ++i) d += __expf(lrelu(f1s[i] + fj) - mx);
  mcol[(size_t)b * NNODES + j] = mx;
  rcol[(size_t)b * NNODES + j] = 1.0f / d;
}

// ---------------------------------------------------------------------------
// Kernel 4: h_prime[b,i,:] = sum_j P(i,j) * Wh[b,j,:],
//   P(i,j) = exp(lrelu(f1_i+f2_j) - m_j) * r_j  generated in-register per
//   16x4 A fragment.  Wh j-chunks double-buffered in LDS and filled with
//   GLOBAL_LOAD_ASYNC_TO_LDS_B128, overlapped with the WMMA K-loop.
// grid = (16 row-tiles, 16 batches), block = 256 (8 waves)
// ---------------------------------------------------------------------------
__global__ __launch_bounds__(256) void k_attn_gemm(const float* __restrict__ Wh,
                                                   const float* __restrict__ f1,
                                                   const float* __restrict__ f2,
                                                   const float* __restrict__ mcol,
                                                   const float* __restrict__ rcol,
                                                   float* __restrict__ out) {
  constexpr int CH = 64;           // j-chunk (K) staged per iteration
  constexpr int LDW = FDIM + 4;    // 132 floats = 528 B (16B aligned, bank-skewed)
  constexpr int NC = NNODES / CH;  // 32 chunks
  __shared__ float WhS[2][CH * LDW];  // 2 x ~33.8 KB
  __shared__ float f2s[2][CH], ms[2][CH], rs[2][CH];
  __shared__ float f1s[128];

  const int tid = threadIdx.x;
  const int b = blockIdx.y;
  const int wave = tid >> 5, lane = tid & 31;
  const int ln = lane & 15, hi = lane >> 4;
  const int rowBase = blockIdx.x * 128;

  const float* WhB = Wh + (size_t)b * NNODES * FDIM;
  const float* f2B = f2 + (size_t)b * NNODES;
  const float* mB = mcol + (size_t)b * NNODES;
  const float* rB = rcol + (size_t)b * NNODES;

  // Issue async stage of chunk 0 into buffer 0 (8 x b128 per thread).
  auto stage = [&](int buf, int j0) {
    const uint32_t base = lds_byte_off(&WhS[buf][0]);
#pragma unroll
    for (int t = 0; t < 8; ++t) {
      int e = (tid + t * 256) * 4;
      int rr = e / FDIM, cc = e % FDIM;
      async_copy_b128(base + (uint32_t)(rr * LDW + cc) * 4u,
                      &WhB[(size_t)(j0 + rr) * FDIM + cc]);
    }
    if (tid < CH) {
      async_copy_b32(lds_byte_off(&f2s[buf][tid]), &f2B[j0 + tid]);
      async_copy_b32(lds_byte_off(&ms[buf][tid]), &mB[j0 + tid]);
      async_copy_b32(lds_byte_off(&rs[buf][tid]), &rB[j0 + tid]);
    }
  };

  stage(0, 0);
  if (tid < 128) f1s[tid] = f1[(size_t)b * NNODES + rowBase + tid];
  __syncthreads();  // publish f1s
  const float f1v = f1s[wave * 16 + ln];  // this lane's row value (M = ln)

  v8f acc[8] = {};

  for (int c = 0; c < NC; ++c) {
    const int cur = c & 1;
    wait_async0();    // this wave's async copies for chunk c landed
    __syncthreads();  // ...and every other wave's too; prev compute done
    if (c + 1 < NC) stage(1 - cur, (c + 1) * CH);  // overlap next copy

    const float* whc = &WhS[cur][0];
    const float* f2c = &f2s[cur][0];
    const float* mc = &ms[cur][0];
    const float* rc = &rs[cur][0];
#pragma unroll 4
    for (int kk = 0; kk < CH; kk += 4) {
      // On-the-fly attention A-fragment (16x4): this lane covers K = kk+hi*2 ..+1
      const int jj = kk + hi * 2;
      float x0 = lrelu(f1v + f2c[jj]);
      float x1 = lrelu(f1v + f2c[jj + 1]);
      v2f afrag;
      afrag.x = __expf(x0 - mc[jj]) * rc[jj];
      afrag.y = __expf(x1 - mc[jj + 1]) * rc[jj + 1];
#pragma unroll
      for (int q = 0; q < 8; ++q) {
        v2f bfrag;
        bfrag.x = whc[(jj + 0) * LDW + q * 16 + ln];
        bfrag.y = whc[(jj + 1) * LDW + q * 16 + ln];
        acc[q] = wmma_f32_16x16x4(afrag, bfrag, acc[q]);
      }
    }
  }

  float* ob = out + (size_t)b * NNODES * FDIM;
#pragma unroll
  for (int q = 0; q < 8; ++q)
#pragma unroll
    for (int v = 0; v < 8; ++v)
      ob[(size_t)(rowBase + wave * 16 + v + hi * 8) * FDIM + q * 16 + ln] = acc[q][v];
}

// ---------------------------------------------------------------------------
extern "C" void kernel_launch(void* const* d_in, const int* in_sizes, int n_in,
                              void* d_out, int out_size, void* d_ws, size_t ws_size,
                              hipStream_t stream) {
  const float* h = (const float*)d_in[0];
  // d_in[1] = adj : unused by the reference computation
  const float* W = (const float*)d_in[2];
  const float* a = (const float*)d_in[3];
  float* out = (float*)d_out;

  float* ws = (float*)d_ws;
  float* Wh = ws;                                   // 16*2048*128 f32 (16 MB)
  float* f1 = Wh + (size_t)NBATCH * NNODES * FDIM;  // 32768
  float* f2 = f1 + NBATCH * NNODES;                 // 32768
  float* mc = f2 + NBATCH * NNODES;                 // 32768
  float* rc = mc + NBATCH * NNODES;                 // 32768

  k_wh_gemm<<<dim3(NNODES / 128, NBATCH), 256, 0, stream>>>(h, W, Wh);
  k_f1f2<<<(NBATCH * NNODES) / 256, 256, 0, stream>>>(Wh, a, f1, f2);
  k_softmax_stats<<<dim3(NNODES / 256, NBATCH), 256, 0, stream>>>(f1, f2, mc, rc);
  k_attn_gemm<<<dim3(NNODES / 128, NBATCH), 256, 0, stream>>>(Wh, f1, f2, mc, rc, out);
}